// GCN_84301618085975
// MI455X (gfx1250) — compile-verified
//
#include <hip/hip_runtime.h>

#define N_NODES 50000
#define N_EDGES 800000
#define NFEAT 512
#define NHID  256
#define NCLASS 64

typedef float v2f __attribute__((ext_vector_type(2)));
typedef float v4f __attribute__((ext_vector_type(4)));
typedef float v8f __attribute__((ext_vector_type(8)));

// ---------------------------------------------------------------------------
// WMMA f32 GEMM: C[M,N] = A[M,K] * B[K,N].
// Block = 128 threads = 4 waves. Flat job = mt * (N/64) + ng, ng fastest, so
// for N=256 all 4 waves of a block share one 16-row A tile (A hits L1 after
// the first wave), and for N=64 the 4 waves share the whole B (L1-resident).
//
// Each wave computes 16 rows x 64 cols with 4 f32 accumulators. The four
// 16-col WMMA tiles are column-interleaved with stride 4:
//     tile t, tile-col n  <->  memory col nb + 4n + t
// so each lane's 4 B elements per k-row are contiguous (one b128 load) and
// each lane's 4 C outputs per row are contiguous (one b128 store).
//
// Lane layouts per CDNA5 ISA 7.12.2 (V_WMMA_F32_16X16X4_F32):
//   A (16x4):  lane = (kHalf<<4)|m,  v2f = A[m][2*kHalf + {0,1}]
//   B (4x16):  lane = (kHalf<<4)|n,  v2f = B[2*kHalf + {0,1}][col(n)]
//   C (16x16): VGPR i: lanes 0-15 -> M=i, lanes 16-31 -> M=i+8, N = lane&15
// ---------------------------------------------------------------------------
template<int K, int N>
__global__ __launch_bounds__(128)
void gemm_wmma_f32(const float* __restrict__ A, const float* __restrict__ B,
                   float* __restrict__ C, int Mtiles) {
  constexpr int NG = N / 64;               // 64-col groups
  const int wave = threadIdx.x >> 5;
  const int lane = threadIdx.x & 31;
  const int job  = blockIdx.x * 4 + wave;
  if (job >= Mtiles * NG) return;          // wave-uniform: EXEC stays all-1s
  const int mt = job / NG;
  const int ng = job % NG;

  const int half = lane >> 4;              // which k-pair of the k=4 slice
  const int idx  = lane & 15;              // A row / B-tile col within tile
  const int nb   = ng * 64;

  v8f acc0 = {}, acc1 = {}, acc2 = {}, acc3 = {};

  const float* __restrict__ arow  = A + (size_t)(mt * 16 + idx) * K;
  const float* __restrict__ bbase = B + nb + 4 * idx;

  for (int k0 = 0; k0 < K; k0 += 4) {
    const int ka = k0 + half * 2;
    v2f a  = *(const v2f*)(arow + ka);
    v4f r0 = *(const v4f*)(bbase + (size_t)ka * N);        // B[ka][4idx+0..3]
    v4f r1 = *(const v4f*)(bbase + (size_t)(ka + 1) * N);  // B[ka+1][...]
    v2f b;
    b.x = r0.x; b.y = r1.x;
    acc0 = __builtin_amdgcn_wmma_f32_16x16x4_f32(false, a, false, b, (short)0,
                                                 acc0, false, false);
    b.x = r0.y; b.y = r1.y;
    acc1 = __builtin_amdgcn_wmma_f32_16x16x4_f32(false, a, false, b, (short)0,
                                                 acc1, false, false);
    b.x = r0.z; b.y = r1.z;
    acc2 = __builtin_amdgcn_wmma_f32_16x16x4_f32(false, a, false, b, (short)0,
                                                 acc2, false, false);
    b.x = r0.w; b.y = r1.w;
    acc3 = __builtin_amdgcn_wmma_f32_16x16x4_f32(false, a, false, b, (short)0,
                                                 acc3, false, false);
  }

  const int mbase = mt * 16 + half * 8;
  float* __restrict__ cptr = C + (size_t)mbase * N + nb + 4 * idx;
#pragma unroll
  for (int i = 0; i < 8; ++i) {
    v4f st;                                 // cols 4*idx + {0,1,2,3} = tiles 0..3
    st.x = acc0[i]; st.y = acc1[i]; st.z = acc2[i]; st.w = acc3[i];
    *(v4f*)(cptr + (size_t)i * N) = st;
  }
}

// ---------------------------------------------------------------------------
// COO SpMM scatter: out[dst[e], :] += w[e] * sup[src[e], :]
// One block per edge, F threads. unsafeAtomicAdd -> global_atomic_add_f32
// (L2 atomic units; support matrix is L2-resident at 51 MB / 192 MB L2).
// ---------------------------------------------------------------------------
template<int F>
__global__ void spmm_scatter(const int* __restrict__ src,
                             const int* __restrict__ dst,
                             const float* __restrict__ w,
                             const float* __restrict__ sup,
                             float* __restrict__ out) {
  const int e = blockIdx.x;
  const int t = threadIdx.x;
  const int s = src[e];
  const int d = dst[e];
  const float wt = w[e];
  const float v = wt * sup[(size_t)s * F + t];
  unsafeAtomicAdd(out + (size_t)d * F + t, v);
}

__global__ void fill_zero(float* __restrict__ p, int n) {
  int i = blockIdx.x * blockDim.x + threadIdx.x;
  const int stride = gridDim.x * blockDim.x;
  for (; i < n; i += stride) p[i] = 0.0f;
}

// h = relu(h + b1[col]); F must be power of two.
__global__ void bias_relu(float* __restrict__ h, const float* __restrict__ b,
                          int total, int fmask) {
  int i = blockIdx.x * blockDim.x + threadIdx.x;
  const int stride = gridDim.x * blockDim.x;
  for (; i < total; i += stride) {
    float v = h[i] + b[i & fmask];
    h[i] = fmaxf(v, 0.0f);
  }
}

// In-place per-row log_softmax over NCLASS=64 with bias add.
__global__ __launch_bounds__(NCLASS)
void logsoftmax_bias(float* __restrict__ out, const float* __restrict__ b2) {
  __shared__ float red[NCLASS];
  const int r = blockIdx.x;
  const int t = threadIdx.x;
  const float v = out[(size_t)r * NCLASS + t] + b2[t];

  red[t] = v;
  __syncthreads();
#pragma unroll
  for (int s = NCLASS / 2; s > 0; s >>= 1) {
    if (t < s) red[t] = fmaxf(red[t], red[t + s]);
    __syncthreads();
  }
  const float mx = red[0];
  __syncthreads();

  red[t] = __expf(v - mx);
  __syncthreads();
#pragma unroll
  for (int s = NCLASS / 2; s > 0; s >>= 1) {
    if (t < s) red[t] += red[t + s];
    __syncthreads();
  }
  const float lse = __logf(red[0]);
  out[(size_t)r * NCLASS + t] = v - mx - lse;
}

// ---------------------------------------------------------------------------
// kernel_launch: inputs in setup_inputs() order:
//   0:x 1:W1 2:b1 3:W2 4:b2 5:edge_weight 6:edge_src 7:edge_dst
// d_out: 50000 x 64 f32 log-probabilities.
// ---------------------------------------------------------------------------
extern "C" void kernel_launch(void* const* d_in, const int* in_sizes, int n_in,
                              void* d_out, int out_size, void* d_ws, size_t ws_size,
                              hipStream_t stream) {
  (void)in_sizes; (void)n_in; (void)out_size; (void)ws_size;

  const float* x  = (const float*)d_in[0];
  const float* W1 = (const float*)d_in[1];
  const float* b1 = (const float*)d_in[2];
  const float* W2 = (const float*)d_in[3];
  const float* b2 = (const float*)d_in[4];
  const float* ew = (const float*)d_in[5];
  const int*   es = (const int*)d_in[6];
  const int*   ed = (const int*)d_in[7];
  float* out = (float*)d_out;

  char* ws = (char*)d_ws;
  float* sup1 = (float*)ws;                                              // 50000*256
  float* h    = (float*)(ws + (size_t)N_NODES * NHID * sizeof(float));   // 50000*256
  float* sup2 = (float*)(ws + (size_t)2 * N_NODES * NHID * sizeof(float)); // 50000*64

  const int Mtiles = N_NODES / 16;  // 3125

  // zero scatter destinations
  fill_zero<<<2048, 256, 0, stream>>>(h, N_NODES * NHID);
  fill_zero<<<1024, 256, 0, stream>>>(out, N_NODES * NCLASS);

  // layer 1: sup1 = x @ W1  (WMMA f32 16x16x4; 4 waves/block share the A tile)
  {
    const int jobs = Mtiles * (NHID / 64);        // 12500
    gemm_wmma_f32<NFEAT, NHID>
        <<<(jobs + 3) / 4, 128, 0, stream>>>(x, W1, sup1, Mtiles);
  }
  // h = SpMM(adj, sup1)
  spmm_scatter<NHID><<<N_EDGES, NHID, 0, stream>>>(es, ed, ew, sup1, h);
  // h = relu(h + b1)
  bias_relu<<<2048, 256, 0, stream>>>(h, b1, N_NODES * NHID, NHID - 1);

  // layer 2: sup2 = h @ W2  (4 waves/block share W2 from L1)
  {
    const int jobs = Mtiles * (NCLASS / 64);      // 3125
    gemm_wmma_f32<NHID, NCLASS>
        <<<(jobs + 3) / 4, 128, 0, stream>>>(h, W2, sup2, Mtiles);
  }
  // out = SpMM(adj, sup2)
  spmm_scatter<NCLASS><<<N_EDGES, NCLASS, 0, stream>>>(es, ed, ew, sup2, out);
  // out = log_softmax(out + b2)
  logsoftmax_bias<<<N_NODES, NCLASS, 0, stream>>>(out, b2);
}